// MoE_GNN_53163105190601
// MI455X (gfx1250) — compile-verified
//
#include <hip/hip_runtime.h>
#include <hip/hip_bf16.h>

// ---- problem constants (match reference) ----
#define NN   32768      // nodes
#define EE   262144     // edges
#define BB   512        // graphs
#define DD   64         // hidden dim
#define DEE  16         // edge attr dim
#define NEXP 8
#define NODES_PER_GRAPH 64

typedef __attribute__((ext_vector_type(2))) float v2f;
typedef __attribute__((ext_vector_type(8))) float v8f;

__device__ __forceinline__ v8f wmma_f32_k4(v2f a, v2f b, v8f c) {
    // D(16x16,f32) = A(16x4,f32) * B(4x16,f32) + C
    return __builtin_amdgcn_wmma_f32_16x16x4_f32(
        /*neg_a=*/false, a, /*neg_b=*/false, b,
        /*c_mod=*/(short)0, c, /*reuse_a=*/false, /*reuse_b=*/false);
}

// ---------------------------------------------------------------------------
// Edge kernel: one wave (32 lanes) per 16-edge tile.
//   proj[16x64] = edge_attr_tile[16x16] @ We[16x64]   (16 WMMA f32 16x16x4)
//   then agg[dst[e]][:] += x[src[e]][:] * proj[e][:]  (hardware f32 atomics)
// ---------------------------------------------------------------------------
__global__ __launch_bounds__(32) void edge_kernel(
    const float* __restrict__ edge_attr,   // [E,16]
    const float* __restrict__ We,          // [16,64]
    const float* __restrict__ x,           // [N,64]
    const int*   __restrict__ esrc,        // [E]
    const int*   __restrict__ edst,        // [E]
    float*       __restrict__ agg)         // [N,64]
{
    __shared__ float att[16 * 16];
    __shared__ int   ssh[16];
    __shared__ int   dsh[16];

    const int tid = threadIdx.x;             // 0..31
    const int e0  = blockIdx.x * 16;

    // 16 edges x 16 attrs = 256 contiguous floats = 64 float4
    const float4* g4 = (const float4*)(edge_attr + (size_t)e0 * DEE);
    float4* a4 = (float4*)att;
    a4[tid]      = g4[tid];
    a4[tid + 32] = g4[tid + 32];
    if (tid < 16) ssh[tid]      = esrc[e0 + tid];
    else          dsh[tid - 16] = edst[e0 + tid - 16];
    __syncthreads();

    const int lo   = tid & 15;
    const int hi   = tid >> 4;      // 0: K rows {0,1}; 1: K rows {2,3}
    const int koff = hi * 2;

    v8f c[4] = {v8f{}, v8f{}, v8f{}, v8f{}};
#pragma unroll
    for (int kc = 0; kc < 4; ++kc) {
        const int kb = kc * 4 + koff;
        v2f a;                       // A 16x4 f32 layout (ISA 7.12.2)
        a.x = att[lo * DEE + kb];
        a.y = att[lo * DEE + kb + 1];
#pragma unroll
        for (int t = 0; t < 4; ++t) {
            const int n = lo + 16 * t;
            v2f b;                   // B 4x16: row striped across lanes
            b.x = We[(size_t)kb * DD + n];
            b.y = We[(size_t)(kb + 1) * DD + n];
            c[t] = wmma_f32_k4(a, b, c[t]);
        }
    }

    // C layout: VGPR i, lanes 0-15 -> row i, lanes 16-31 -> row i+8; col = lo+16t
#pragma unroll
    for (int i = 0; i < 8; ++i) {
        const int m = i + 8 * hi;
        const int s = ssh[m];
        const int d = dsh[m];
        const float* xr = x   + (size_t)s * DD;
        float*       ar = agg + (size_t)d * DD;
#pragma unroll
        for (int t = 0; t < 4; ++t) {
            const int n = lo + 16 * t;
            unsafeAtomicAdd(ar + n, xr[n] * c[t][i]);
        }
    }
}

// ---------------------------------------------------------------------------
// Node kernel: one block = one graph (64 contiguous nodes), 4 waves.
//   h = relu(x@W1 + agg@W2 + b)  (128 WMMA per wave), then mean-pool,
//   then either store graph_emb (embedding pass) or emb@Wout+bout (expert).
// ---------------------------------------------------------------------------
__global__ __launch_bounds__(128) void node_kernel(
    const float* __restrict__ x,          // [N,64]
    const float* __restrict__ agg,        // [N,64]
    const float* __restrict__ W1,         // [64,64]
    const float* __restrict__ W2,         // [64,64]
    const float* __restrict__ bvec,       // [64]
    const float* __restrict__ Wout,       // [64] or null
    const float* __restrict__ bout,       // [1]  or null
    const int*   __restrict__ batch,      // [N]
    float*       __restrict__ graph_emb,  // [B,64]
    float*       __restrict__ per_graph,  // [B,NEXP]
    int expert)                           // -1 = embedding pass
{
    __shared__ float xs[64 * 64];
    __shared__ float as_[64 * 64];
    __shared__ float red[4 * 64];
    __shared__ float emb_s[64];

    const int tid = threadIdx.x;
    const int n0  = blockIdx.x * NODES_PER_GRAPH;

    // load two 64x64 tiles (1024 float4 each), 8 float4 per thread per tile
    const float4* xg = (const float4*)(x   + (size_t)n0 * DD);
    const float4* ag = (const float4*)(agg + (size_t)n0 * DD);
    float4* xs4 = (float4*)xs;
    float4* as4 = (float4*)as_;
#pragma unroll
    for (int j = 0; j < 8; ++j) {
        xs4[tid + 128 * j] = xg[tid + 128 * j];
        as4[tid + 128 * j] = ag[tid + 128 * j];
    }
    __syncthreads();

    const int w    = tid >> 5;     // wave id: M-tile row block
    const int l    = tid & 31;
    const int lo   = l & 15;
    const int hi   = l >> 4;
    const int koff = hi * 2;
    const int mrow = w * 16 + lo;

    v8f c[4] = {v8f{}, v8f{}, v8f{}, v8f{}};
#pragma unroll
    for (int kc = 0; kc < 16; ++kc) {
        const int kb = kc * 4 + koff;
        v2f a1, a2;
        a1.x = xs[mrow * DD + kb];  a1.y = xs[mrow * DD + kb + 1];
        a2.x = as_[mrow * DD + kb]; a2.y = as_[mrow * DD + kb + 1];
#pragma unroll
        for (int t = 0; t < 4; ++t) {
            const int n = lo + 16 * t;
            v2f b1, b2;
            b1.x = W1[(size_t)kb * DD + n]; b1.y = W1[(size_t)(kb + 1) * DD + n];
            c[t] = wmma_f32_k4(a1, b1, c[t]);
            b2.x = W2[(size_t)kb * DD + n]; b2.y = W2[(size_t)(kb + 1) * DD + n];
            c[t] = wmma_f32_k4(a2, b2, c[t]);
        }
    }

    // bias + relu + column sums (pooling). Lane holds 8 rows of its M-tile.
#pragma unroll
    for (int t = 0; t < 4; ++t) {
        const int n  = lo + 16 * t;
        const float bn = bvec[n];
        float s = 0.f;
#pragma unroll
        for (int i = 0; i < 8; ++i) {
            float h = c[t][i] + bn;
            s += fmaxf(h, 0.f);
        }
        s += __shfl_xor(s, 16, 32);      // combine row halves (M=i and M=i+8)
        if (hi == 0) red[w * 64 + n] = s;
    }
    __syncthreads();

    if (tid < 64) {
        const float e = (red[tid] + red[64 + tid] + red[128 + tid] + red[192 + tid])
                        * (1.0f / (float)NODES_PER_GRAPH);
        emb_s[tid] = e;
    }
    __syncthreads();

    const int gid = batch[n0];
    if (expert < 0) {
        if (tid < 64) graph_emb[(size_t)gid * DD + tid] = emb_s[tid];
    } else {
        if (tid < 64) red[tid] = emb_s[tid] * Wout[tid];
        __syncthreads();
        if (tid == 0) {
            float s = bout[0];
#pragma unroll
            for (int j = 0; j < 64; ++j) s += red[j];
            per_graph[(size_t)gid * NEXP + expert] = s;
        }
    }
}

// ---------------------------------------------------------------------------
// Gating + top-2 + combine. One thread per graph.
// out layout: [0,512) final_output, [512, 512+512*8) routing_weights
// ---------------------------------------------------------------------------
__global__ __launch_bounds__(128) void gate_combine_kernel(
    const float* __restrict__ graph_emb,  // [B,64]
    const float* __restrict__ Wg,         // [64,8]
    const float* __restrict__ bg,         // [8]
    const float* __restrict__ per_graph,  // [B,8]
    float*       __restrict__ out)
{
    const int g = blockIdx.x * blockDim.x + threadIdx.x;
    if (g >= BB) return;
    const float* e = graph_emb + (size_t)g * DD;

    float logits[NEXP];
#pragma unroll
    for (int j = 0; j < NEXP; ++j) logits[j] = bg[j];
    for (int d = 0; d < DD; ++d) {
        const float ev = e[d];
        const float* wr = Wg + (size_t)d * NEXP;
#pragma unroll
        for (int j = 0; j < NEXP; ++j) logits[j] += ev * wr[j];
    }
    float mx = logits[0];
#pragma unroll
    for (int j = 1; j < NEXP; ++j) mx = fmaxf(mx, logits[j]);
    float p[NEXP], den = 0.f;
#pragma unroll
    for (int j = 0; j < NEXP; ++j) { p[j] = __expf(logits[j] - mx); den += p[j]; }
    const float inv = 1.0f / den;
#pragma unroll
    for (int j = 0; j < NEXP; ++j) {
        p[j] *= inv;
        out[BB + (size_t)g * NEXP + j] = p[j];     // routing_weights
    }
    // top-2 (ties -> lowest index, like lax.top_k)
    int i0 = 0;
#pragma unroll
    for (int j = 1; j < NEXP; ++j) if (p[j] > p[i0]) i0 = j;
    int i1 = (i0 == 0) ? 1 : 0;
#pragma unroll
    for (int j = 0; j < NEXP; ++j) if (j != i0 && p[j] > p[i1]) i1 = j;
    // softmax over the two selected probabilities
    const float m2 = fmaxf(p[i0], p[i1]);
    const float a0 = __expf(p[i0] - m2), a1 = __expf(p[i1] - m2);
    const float w0 = a0 / (a0 + a1), w1 = a1 / (a0 + a1);
    out[g] = w0 * per_graph[(size_t)g * NEXP + i0]
           + w1 * per_graph[(size_t)g * NEXP + i1];
}

// ---------------------------------------------------------------------------
extern "C" void kernel_launch(void* const* d_in, const int* in_sizes, int n_in,
                              void* d_out, int out_size, void* d_ws, size_t ws_size,
                              hipStream_t stream) {
    const float* x         = (const float*)d_in[0];
    const float* edge_attr = (const float*)d_in[1];
    const float* emb_We    = (const float*)d_in[2];
    const float* emb_W1    = (const float*)d_in[3];
    const float* emb_W2    = (const float*)d_in[4];
    const float* emb_b     = (const float*)d_in[5];
    const float* Wg        = (const float*)d_in[6];
    const float* bg        = (const float*)d_in[7];
    const float* ex_We     = (const float*)d_in[8];
    const float* ex_W1     = (const float*)d_in[9];
    const float* ex_W2     = (const float*)d_in[10];
    const float* ex_b      = (const float*)d_in[11];
    const float* ex_Wout   = (const float*)d_in[12];
    const float* ex_bout   = (const float*)d_in[13];
    const int*   esrc      = (const int*)d_in[14];
    const int*   edst      = (const int*)d_in[15];
    const int*   batch     = (const int*)d_in[16];

    float* agg       = (float*)d_ws;                       // N*D
    float* graph_emb = agg + (size_t)NN * DD;              // B*D
    float* per_graph = graph_emb + (size_t)BB * DD;        // B*NEXP
    float* out       = (float*)d_out;

    for (int p = 0; p < 1 + NEXP; ++p) {
        const int expert = p - 1;
        const float *We, *W1, *W2, *bb, *Wout = nullptr, *bout = nullptr;
        if (p == 0) {
            We = emb_We; W1 = emb_W1; W2 = emb_W2; bb = emb_b;
        } else {
            We   = ex_We   + (size_t)expert * DEE * DD;
            W1   = ex_W1   + (size_t)expert * DD * DD;
            W2   = ex_W2   + (size_t)expert * DD * DD;
            bb   = ex_b    + (size_t)expert * DD;
            Wout = ex_Wout + (size_t)expert * DD;
            bout = ex_bout + expert;
        }
        hipMemsetAsync(agg, 0, (size_t)NN * DD * sizeof(float), stream);
        edge_kernel<<<EE / 16, 32, 0, stream>>>(edge_attr, We, x, esrc, edst, agg);
        node_kernel<<<BB, 128, 0, stream>>>(x, agg, W1, W2, bb, Wout, bout,
                                            batch, graph_emb, per_graph, expert);
    }
    gate_combine_kernel<<<(BB + 127) / 128, 128, 0, stream>>>(
        graph_emb, Wg, bg, per_graph, out);
}